// EPrompt_9234179687675
// MI455X (gfx1250) — compile-verified
//
#include <hip/hip_runtime.h>
#include <hip/hip_bf16.h>

typedef __attribute__((ext_vector_type(2))) float v2f;
typedef __attribute__((ext_vector_type(8))) float v8f;
typedef __attribute__((ext_vector_type(4))) float f4;

#define TOKENS 197
#define DIM    768
#define BATCH  128
#define POOL   100
#define POOLP  112           // padded pool rows for WMMA N tiles (7*16)
#define TOPK   5
#define CHUNK  15360         // 20*12*64 floats per gathered chunk
#define NCHUNK 6400          // 5*2*128*5
#define OUT_PROMPT_ELEMS 98304000ull

// ---------------- 1) mean over tokens + L2 normalize: x_norm (128,768) -------
__global__ void ep_mean_norm(const float* __restrict__ x, float* __restrict__ xn) {
    const int b = blockIdx.x;          // 0..127
    const int tid = threadIdx.x;       // 0..255, 3 dims each
    const float* xb = x + (size_t)b * TOKENS * DIM;
    float a0 = 0.f, a1 = 0.f, a2 = 0.f;
    for (int t = 0; t < TOKENS; ++t) {
        const float* row = xb + t * DIM;
        a0 += row[tid];
        a1 += row[tid + 256];
        a2 += row[tid + 512];
    }
    const float inv = 1.0f / (float)TOKENS;
    float m0 = a0 * inv, m1 = a1 * inv, m2 = a2 * inv;

    __shared__ float red[256];
    red[tid] = m0 * m0 + m1 * m1 + m2 * m2;
    __syncthreads();
    for (int s = 128; s > 0; s >>= 1) {
        if (tid < s) red[tid] += red[tid + s];
        __syncthreads();
    }
    float r = rsqrtf(fmaxf(red[0], 1e-12f));
    float* o = xn + (size_t)b * DIM;
    o[tid] = m0 * r; o[tid + 256] = m1 * r; o[tid + 512] = m2 * r;
}

// ---------------- 2) L2 normalize prompt_key -> pk_norm (112,768), rows>=100 zero
__global__ void ep_key_norm(const float* __restrict__ pk, float* __restrict__ pkn) {
    const int p = blockIdx.x;          // 0..111
    const int tid = threadIdx.x;       // 0..255
    float* o = pkn + (size_t)p * DIM;
    if (p >= POOL) {                   // zero padding rows so WMMA tiles need no guard
        o[tid] = 0.f; o[tid + 256] = 0.f; o[tid + 512] = 0.f;
        return;
    }
    const float* row = pk + (size_t)p * DIM;
    float m0 = row[tid], m1 = row[tid + 256], m2 = row[tid + 512];
    __shared__ float red[256];
    red[tid] = m0 * m0 + m1 * m1 + m2 * m2;
    __syncthreads();
    for (int s = 128; s > 0; s >>= 1) {
        if (tid < s) red[tid] += red[tid + s];
        __syncthreads();
    }
    float r = rsqrtf(fmaxf(red[0], 1e-12f));
    o[tid] = m0 * r; o[tid + 256] = m1 * r; o[tid + 512] = m2 * r;
}

// ---------------- 3) similarity = x_norm @ pk_normT via V_WMMA_F32_16X16X4_F32
// grid (8,7), one wave (32 threads) per 16x16 C tile; sim is (128,112)
__global__ void ep_sim_wmma(const float* __restrict__ xn,
                            const float* __restrict__ pkn,
                            float* __restrict__ sim) {
    const int lane = threadIdx.x;      // 0..31
    const int half = lane >> 4;        // K sub-pair selector
    const int lr   = lane & 15;        // row (A) / col (B) within tile
    const int row0 = blockIdx.x * 16;
    const int col0 = blockIdx.y * 16;

    const float* arow = xn  + (size_t)(row0 + lr) * DIM;  // A: M = lr
    const float* brow = pkn + (size_t)(col0 + lr) * DIM;  // B: N = lr (B=pkT)

    v8f c = {};
    for (int k0 = 0; k0 < DIM; k0 += 4) {
        const int kk = k0 + (half << 1);   // lanes 0-15: K=k0,k0+1 ; 16-31: K=k0+2,k0+3
        v2f a  = *(const v2f*)(arow + kk);
        v2f bb = *(const v2f*)(brow + kk);
        c = __builtin_amdgcn_wmma_f32_16x16x4_f32(
                /*neg_a=*/false, a, /*neg_b=*/false, bb,
                /*c_mod=*/(short)0, c, /*reuse_a=*/false, /*reuse_b=*/false);
    }
    const int col = col0 + lr;
#pragma unroll
    for (int r = 0; r < 8; ++r) {
        const int row = row0 + r + (half << 3);  // lanes 16-31 hold M = 8..15
        sim[(size_t)row * POOLP + col] = c[r];
    }
}

// ---------------- 4) top-5 per row (tie -> lower index), plus per-row top-k sum
__global__ void ep_topk(const float* __restrict__ sim, int* __restrict__ idx,
                        float* __restrict__ rowsum) {
    const int b = threadIdx.x;         // one block of 128
    const float* s = sim + (size_t)b * POOLP;
    int chosen[TOPK];
    float total = 0.f;
    for (int k = 0; k < TOPK; ++k) {
        float best = -3.0e38f;
        int bi = 0;
        for (int p = 0; p < POOL; ++p) {
            bool skip = false;
            for (int j = 0; j < k; ++j) skip = skip || (chosen[j] == p);
            const float v = s[p];
            if (!skip && v > best) { best = v; bi = p; }
        }
        chosen[k] = bi;
        idx[b * TOPK + k] = bi;
        total += best;
    }
    rowsum[b] = total;
}

// ---------------- 5) reduce_sim = sum(rowsum)/128 -> d_out tail element --------
__global__ void ep_reduce(const float* __restrict__ rowsum, float* __restrict__ out_scalar) {
    __shared__ float red[128];
    const int t = threadIdx.x;
    red[t] = rowsum[t];
    __syncthreads();
    for (int s = 64; s > 0; s >>= 1) {
        if (t < s) red[t] += red[t + s];
        __syncthreads();
    }
    if (t == 0) out_scalar[0] = red[0] / (float)BATCH;
}

// ---------------- 6) the big gather: prompt[:, :, idx] -> out (393 MB of stores)
// One block per (l,d,b,k) chunk of 15360 floats; b128 loads (L2-resident source),
// non-temporal b128 stores (write-once output, keep it out of L2).
__global__ void ep_gather(const float* __restrict__ prompt, const int* __restrict__ idx,
                          float* __restrict__ out) {
    const int lin = blockIdx.x;        // 0..6399 == ((l*2+d)*128 + b)*5 + k
    const int k  = lin % TOPK;
    const int b  = (lin / TOPK) % BATCH;
    const int ld = lin / (TOPK * BATCH);   // l*2+d in 0..9
    const int p  = idx[b * TOPK + k];      // scalar (uniform per block)

    const f4* __restrict__ src = (const f4*)(prompt + ((size_t)ld * POOL + p) * CHUNK);
    f4* __restrict__ dst = (f4*)(out + (size_t)lin * CHUNK);
    const int t = threadIdx.x;             // 256 threads, 15 f4 each (3840 total)
#pragma unroll
    for (int i = 0; i < 15; ++i) {
        __builtin_nontemporal_store(src[t + i * 256], dst + t + i * 256);
    }
}

extern "C" void kernel_launch(void* const* d_in, const int* in_sizes, int n_in,
                              void* d_out, int out_size, void* d_ws, size_t ws_size,
                              hipStream_t stream) {
    const float* x_embed    = (const float*)d_in[0];   // (128,197,768)
    const float* prompt     = (const float*)d_in[1];   // (5,2,100,20,12,64)
    const float* prompt_key = (const float*)d_in[2];   // (100,768)
    float* out = (float*)d_out;                        // 98304000 floats + 1 scalar

    // workspace layout (floats): ~780 KB total
    float* ws     = (float*)d_ws;
    float* xn     = ws;                                // 128*768   = 98304
    float* pkn    = ws + 98304;                        // 112*768   = 86016
    float* sim    = ws + 184320;                       // 128*112   = 14336
    int*   idx    = (int*)(ws + 198656);               // 128*5     = 640
    float* rowsum = ws + 199296;                       // 128

    ep_mean_norm<<<BATCH, 256, 0, stream>>>(x_embed, xn);
    ep_key_norm<<<POOLP, 256, 0, stream>>>(prompt_key, pkn);
    ep_sim_wmma<<<dim3(BATCH / 16, POOLP / 16), 32, 0, stream>>>(xn, pkn, sim);
    ep_topk<<<1, BATCH, 0, stream>>>(sim, idx, rowsum);
    ep_reduce<<<1, BATCH, 0, stream>>>(rowsum, out + OUT_PROMPT_ELEMS);
    ep_gather<<<NCHUNK, 256, 0, stream>>>(prompt, idx, out);
}